// SparseReservoir_85203561218540
// MI455X (gfx1250) — compile-verified
//
#include <hip/hip_runtime.h>
#include <math.h>

// Echo-state network for MI455X (gfx1250, wave32).
// Phase 1: big parallel GEMM (inputs @ W_in) via v_wmma_f32_16x16x4_f32,
//          2 M-tiles per wave sharing the B fragment, written into d_out.
// Phase 2: 2048 sequential step kernels; each does S_prev @ W_res with f32
//          WMMA (W_res = 4MB, L2-resident across all steps), fused
//          sine-decay activation + leaky integration, in-place on d_out.
// All flat offsets fit in 31 bits -> int indexing to keep address math on
// the scalar pipe / 32-bit VALU.

typedef __attribute__((ext_vector_type(2))) float v2f;
typedef __attribute__((ext_vector_type(8))) float v8f;

#define LEAK_RATE  0.3f
#define SINE_DECAY 0.1f

// ---------------------------------------------------------------------------
// Phase 1: C[M,N] = A[M,K] @ B[K,N].  M=B*T=65536, K=D=256, N=R=1024.
// One wave computes a 32x16 C slab (two 16x16 WMMA tiles stacked in M),
// reusing one B fragment for both -> 2 VMEM per WMMA.
// ---------------------------------------------------------------------------
__global__ __launch_bounds__(256)
void esn_in_gemm(const float* __restrict__ A, const float* __restrict__ Bm,
                 float* __restrict__ C, int M, int N, int K) {
  const int lane = threadIdx.x & 31;
  const int wave = threadIdx.x >> 5;
  const int nt   = blockIdx.x;                     // N/16 tiles
  const int m0   = (blockIdx.y * 8 + wave) * 32;   // 32-row slab
  const int n0   = nt * 16;
  const int hi   = lane >> 4;                      // K-pair select
  const int l15  = lane & 15;

  v8f c0 = {}, c1 = {};
  // A fragment (16x4): lane l15 = row, hi selects K {0,1} vs {2,3}
  const float* arow0 = A  + (m0 + l15) * K + hi * 2;        // rows m0..m0+15
  const float* arow1 = arow0 + 16 * K;                      // rows m0+16..m0+31
  // B fragment (4x16): lane l15 = col, hi selects K rows {0,1} vs {2,3}
  const float* bcol  = Bm + (hi * 2) * N + (n0 + l15);

  #pragma unroll 4
  for (int k = 0; k < K; k += 4) {
    v2f a0 = { arow0[k], arow0[k + 1] };
    v2f a1 = { arow1[k], arow1[k + 1] };
    v2f b  = { bcol[k * N], bcol[(k + 1) * N] };
    c0 = __builtin_amdgcn_wmma_f32_16x16x4_f32(false, a0, false, b,
                                               (short)0, c0, false, false);
    c1 = __builtin_amdgcn_wmma_f32_16x16x4_f32(false, a1, false, b,
                                               (short)0, c1, false, false);
  }

  // C/D layout: VGPR v, lanes 0-15 -> M=v, lanes 16-31 -> M=v+8; N = l15.
  float* crow0 = C + (m0 + hi * 8) * N + n0 + l15;
  float* crow1 = crow0 + 16 * N;
  #pragma unroll
  for (int v = 0; v < 8; ++v) {
    crow0[v * N] = c0[v];
    crow1[v * N] = c1[v];
  }
}

// ---------------------------------------------------------------------------
// Phase 2: one recurrence step.
//   pre   = U_t + S_prev @ W_res        (U_t already in Out[:,t,:])
//   act   = exp(-0.1*|pre|) * sin(pre)
//   S_new = 0.7*S_prev + 0.3*act        -> Out[:,t,:] and compact Snext
// M=32 (batch), N=K=R=1024 -> 2x64 = 128 tiles, one per wave, so the serial
// per-wave chain stays minimal (step latency bound).  16 blocks x 8 waves.
// ---------------------------------------------------------------------------
__global__ __launch_bounds__(256)
void esn_step(const float* __restrict__ Sprev, const float* __restrict__ Wres,
              float* __restrict__ Out, float* __restrict__ Snext,
              int t, int T, int R, int hasPrev) {
  const int lane = threadIdx.x & 31;
  const int g    = blockIdx.x * 8 + (threadIdx.x >> 5);
  const int mt   = g >> 6;          // 0..1
  const int nt   = g & 63;          // 0..63
  const int m0   = mt * 16;
  const int n0   = nt * 16;
  const int hi   = lane >> 4;
  const int l15  = lane & 15;

  v8f c = {};
  if (hasPrev) {
    const float* arow = Sprev + (m0 + l15) * R + hi * 2;
    const float* bcol = Wres  + (hi * 2) * R + (n0 + l15);
    #pragma unroll 4
    for (int k = 0; k < R; k += 4) {
      v2f a = { arow[k], arow[k + 1] };
      v2f b = { bcol[k * R], bcol[(k + 1) * R] };
      c = __builtin_amdgcn_wmma_f32_16x16x4_f32(false, a, false, b,
                                                (short)0, c, false, false);
    }
  }

  const int tR = t * R;
  #pragma unroll
  for (int v = 0; v < 8; ++v) {
    const int row  = m0 + v + hi * 8;            // batch index b
    const int col  = n0 + l15;                   // reservoir index r
    const int oidx = (row * T) * R + tR + col;   // < 2^26, int-safe
    float pre  = c[v] + Out[oidx];               // u_t + recurrent term
    float act  = __expf(-SINE_DECAY * fabsf(pre)) * __sinf(pre);
    float prev = hasPrev ? Sprev[row * R + col] : 0.0f;
    float ns   = (1.0f - LEAK_RATE) * prev + LEAK_RATE * act;
    Out[oidx] = ns;
    Snext[row * R + col] = ns;
  }
}

// ---------------------------------------------------------------------------
extern "C" void kernel_launch(void* const* d_in, const int* in_sizes, int n_in,
                              void* d_out, int out_size, void* d_ws, size_t ws_size,
                              hipStream_t stream) {
  (void)in_sizes; (void)n_in; (void)out_size; (void)ws_size;
  const float* inputs = (const float*)d_in[0];   // [B,T,D]
  const float* W_in   = (const float*)d_in[1];   // [D,R]
  const float* W_res  = (const float*)d_in[2];   // [R,R]
  float* out = (float*)d_out;                    // [B,T,R]

  const int Bb = 32, T = 2048, D = 256, R = 1024;
  const int M = Bb * T;

  // Phase 1: input contributions straight into d_out.
  // Each wave covers 32 rows; 8 waves/block -> 256 rows per block.y.
  dim3 g1(R / 16, M / 256);
  esn_in_gemm<<<g1, dim3(256), 0, stream>>>(inputs, W_in, out, M, R, D);

  // Phase 2: ping-pong compact state in workspace (2 x 32*1024 floats).
  float* s0 = (float*)d_ws;
  float* s1 = s0 + Bb * R;
  for (int t = 0; t < T; ++t) {
    float* sIn  = (t & 1) ? s1 : s0;   // buf[t&1]   (unused at t==0)
    float* sOut = (t & 1) ? s0 : s1;   // buf[(t+1)&1]
    esn_step<<<dim3(16), dim3(256), 0, stream>>>(sIn, W_res, out, sOut,
                                                 t, T, R, t > 0 ? 1 : 0);
  }
}